// Sample_14482629722280
// MI455X (gfx1250) — compile-verified
//
#include <hip/hip_runtime.h>
#include <stdint.h>

// Furthest Point Sampling, MI455X (gfx1250, wave32).
// B=8 batches x G=16 workgroups; each WG keeps its 8192-point chunk
// (x,y,z,dist) entirely in LDS (128 KB of 320 KB/WGP), staged via
// GLOBAL_LOAD_ASYNC_TO_LDS_B128. Each of the 2047 sequential rounds does an
// LDS-only min-update + argmax (float cmp/cndmask per point, u64 key packed
// once per thread), wave32 shuffle reduction, then combines the 16 per-WG
// candidates with one u64 atomicMax in L2 + release/acquire counter barrier.

#define BATCHES 8
#define NPTS    131072
#define NPOINT  2048
#define NWG     16                 // workgroups per batch
#define CH      (NPTS / NWG)       // 8192 points per WG
#define BLOCK   1024               // 32 wave32 waves
#define PPT     (CH / BLOCK)       // 8 points per thread

#if defined(__gfx1250__) && \
    __has_builtin(__builtin_amdgcn_global_load_async_to_lds_b128) && \
    __has_builtin(__builtin_amdgcn_s_wait_asynccnt)
#define ASYNC_LDS 1
#else
#define ASYNC_LDS 0
#endif

typedef int v4i __attribute__((ext_vector_type(4)));

__global__ void fps_init_ws(unsigned long long* cand, unsigned int* cnt) {
    int i = blockIdx.x * blockDim.x + threadIdx.x;
    if (i < BATCHES * NPOINT) { cand[i] = 0ull; cnt[i] = 0u; }
}

__global__ __launch_bounds__(BLOCK)
void fps_kernel(const float* __restrict__ xyz, int* __restrict__ out,
                unsigned long long* __restrict__ cand,
                unsigned int* __restrict__ cnt) {
    __shared__ __attribute__((aligned(16))) float xs[CH];
    __shared__ __attribute__((aligned(16))) float ys[CH];
    __shared__ __attribute__((aligned(16))) float zs[CH];
    __shared__ float dist[CH];
    __shared__ unsigned long long red[BLOCK / 32];
    __shared__ int bcast;

    const int b    = blockIdx.x / NWG;
    const int w    = blockIdx.x % NWG;
    const int tid  = threadIdx.x;
    const int base = w * CH;
    const float* __restrict__ xb = xyz + (size_t)b * 3 * NPTS;

    // ---- Stage this WG's chunk into LDS (async DMA path when available) ----
#if ASYNC_LDS
    {
        const float* srcs[3] = { xb + base, xb + NPTS + base, xb + 2 * NPTS + base };
        float*       dsts[3] = { xs, ys, zs };
        for (int a = 0; a < 3; ++a) {
            #pragma unroll
            for (int k = 0; k < (CH / 4) / BLOCK; ++k) {
                int j = (tid + k * BLOCK) * 4;   // float4 granules, 16B aligned
                __builtin_amdgcn_global_load_async_to_lds_b128(
                    (v4i*)(srcs[a] + j), (v4i*)(dsts[a] + j), 0, 0);
            }
        }
        __builtin_amdgcn_s_wait_asynccnt(0);
    }
#else
    for (int i = tid; i < CH; i += BLOCK) {
        xs[i] = xb[base + i];
        ys[i] = xb[NPTS + base + i];
        zs[i] = xb[2 * NPTS + base + i];
    }
#endif
    for (int i = tid; i < CH; i += BLOCK) dist[i] = 1e10f;
    __syncthreads();

    if (w == 0 && tid == 0) out[b * NPOINT + 0] = 0;   // reference: first idx is 0

    unsigned long long* candB = cand + (size_t)b * NPOINT;
    unsigned int*       cntB  = cnt  + (size_t)b * NPOINT;

    int prev = 0;
    for (int t = 1; t < NPOINT; ++t) {
        // Coordinates of last-picked point (uniform -> scalar loads, L2 hit).
        int pidx = __builtin_amdgcn_readfirstlane(prev);
        float px = xb[pidx];
        float py = xb[NPTS + pidx];
        float pz = xb[2 * NPTS + pidx];

        // LDS-only min-update + per-thread argmax. Per-thread scan order is
        // increasing index, so strict '>' keeps the lowest index among ties,
        // matching jnp.argmax. Key is packed once per thread afterwards.
        float bestd = -1.0f;
        int   besti = 0;
        #pragma unroll
        for (int k = 0; k < PPT; ++k) {
            int i  = tid + k * BLOCK;
            float dx = xs[i] - px, dy = ys[i] - py, dz = zs[i] - pz;
            float d  = dx * dx + dy * dy + dz * dz;
            float dd = dist[i];
            d = d < dd ? d : dd;
            dist[i] = d;
            if (d > bestd) { bestd = d; besti = i; }
        }
        // (dist_bits, ~global_idx): u64 max == argmax, lowest-index tie-break.
        unsigned long long key =
            ((unsigned long long)__float_as_uint(bestd) << 32) |
            (unsigned long long)(unsigned int)~(base + besti);

        // wave32 shuffle reduction, then cross-wave via LDS.
        #pragma unroll
        for (int off = 16; off > 0; off >>= 1) {
            unsigned long long o = __shfl_down(key, (unsigned)off, 32);
            key = o > key ? o : key;
        }
        if ((tid & 31) == 0) red[tid >> 5] = key;
        __syncthreads();
        if (tid < 32) {
            key = red[tid];
            #pragma unroll
            for (int off = 16; off > 0; off >>= 1) {
                unsigned long long o = __shfl_down(key, (unsigned)off, 32);
                key = o > key ? o : key;
            }
            if (tid == 0) {
                // Cross-WG combine: max-candidate + release arrival counter,
                // acquire spin until all NWG chunks contributed.
                __hip_atomic_fetch_max(&candB[t], key,
                                       __ATOMIC_RELAXED, __HIP_MEMORY_SCOPE_AGENT);
                __hip_atomic_fetch_add(&cntB[t], 1u,
                                       __ATOMIC_RELEASE, __HIP_MEMORY_SCOPE_AGENT);
                while (__hip_atomic_load(&cntB[t], __ATOMIC_ACQUIRE,
                                         __HIP_MEMORY_SCOPE_AGENT) < NWG) {
                    __builtin_amdgcn_s_sleep(1);
                }
                unsigned long long wk =
                    __hip_atomic_load(&candB[t], __ATOMIC_RELAXED,
                                      __HIP_MEMORY_SCOPE_AGENT);
                int widx = (int)~(unsigned int)(wk & 0xFFFFFFFFull);
                bcast = widx;
                if (w == 0) out[b * NPOINT + t] = widx;
            }
        }
        __syncthreads();
        prev = bcast;
    }
}

extern "C" void kernel_launch(void* const* d_in, const int* in_sizes, int n_in,
                              void* d_out, int out_size, void* d_ws, size_t ws_size,
                              hipStream_t stream) {
    (void)in_sizes; (void)n_in; (void)out_size; (void)ws_size;
    const float* xyz = (const float*)d_in[0];
    int* out = (int*)d_out;  // int32 indices

    unsigned long long* cand = (unsigned long long*)d_ws;
    unsigned int* cnt =
        (unsigned int*)((char*)d_ws + (size_t)BATCHES * NPOINT * sizeof(unsigned long long));

    int total = BATCHES * NPOINT;
    fps_init_ws<<<(total + 255) / 256, 256, 0, stream>>>(cand, cnt);
    fps_kernel<<<BATCHES * NWG, BLOCK, 0, stream>>>(xyz, out, cand, cnt);
}